// pLayer_45406394253471
// MI455X (gfx1250) — compile-verified
//
#include <hip/hip_runtime.h>
#include <math.h>

#define F_    4
#define V_    4
#define E_    64
#define NIN_  1024
#define NOUT_ 1024
#define M_    1026      // NIN + 2
#define GMIN_ 0.01f
#define GMAX_ 1.0f
#define PGMIN_ 0.1f
// nf = (u*2-1)*0.1 + 1 = 0.2*u + 0.9  (strictly positive)

typedef __attribute__((ext_vector_type(2))) float v2f;
typedef __attribute__((ext_vector_type(8))) float v8f;

__device__ __forceinline__ float clip1(float x) { return fminf(fmaxf(x, -1.f), 1.f); }
__device__ __forceinline__ float clampG(float x) { return fminf(fmaxf(x, -GMAX_), GMAX_); }
__device__ __forceinline__ float frcp(float x) { return __builtin_amdgcn_rcpf(x); }
// sigmoid via hw rcp: 1/(1+exp(-x))
__device__ __forceinline__ float fsigm(float x) { return frcp(1.f + __expf(-x)); }
// INV[sel]: sel=0 -> 1-x ; sel=1 -> sigmoid(4*(0.5-x))  (branchless select)
__device__ __forceinline__ float inv_fn(float x, float sel) {
    float s = fsigm(2.f - 4.f * x);
    return (sel > 0.5f) ? s : (1.f - x);
}
// tanh via hw rcp: 2*sigmoid(2z)-1
__device__ __forceinline__ float ftanh(float z) {
    return 2.f * fsigm(2.f * z) - 1.f;
}

// ---------- K0: per-column pg[n]=PGMIN/min_m|tc|, gtsum[n]=pg*sum_m|tc|, act select ----------
__global__ void k_col(const float* __restrict__ theta, const float* __restrict__ ca,
                      const float* __restrict__ ga, float* __restrict__ pg,
                      float* __restrict__ gtsum, float* __restrict__ actsel) {
    int n = blockIdx.x * blockDim.x + threadIdx.x;
    if (n >= NOUT_) return;
    float mn = 3.4e38f, s = 0.f;
    for (int m = 0; m < M_; ++m) {
        float gi = fabsf(clampG(theta[(size_t)m * NOUT_ + n]));
        mn = fminf(mn, gi);
        s += gi;
    }
    float p = PGMIN_ / mn;   // keep IEEE divide here (one per column; exactness matters)
    pg[n] = p;
    gtsum[n] = p * s;
    float x0 = clip1(ca[n]) + ga[n];
    float x1 = clip1(ca[NOUT_ + n]) + ga[NOUT_ + n];
    actsel[n] = (x1 > x0) ? 1.f : 0.f;   // argmax (tie -> 0)
}

// ---------- K1: beta (negation-branch) hard select per m ----------
__global__ void k_beta(const float* __restrict__ cn, const float* __restrict__ gn,
                       float* __restrict__ betasel) {
    int m = blockIdx.x * blockDim.x + threadIdx.x;
    if (m >= M_) return;
    float x0 = clip1(cn[m]) + gn[m];
    float x1 = clip1(cn[M_ + m]) + gn[M_ + m];
    betasel[m] = (x1 > x0) ? 1.f : 0.f;
}

// ---------- K2: rcolsum[f,v,n] = 1/(sum_m |tz*nf*fm| + 1e-10) ----------
__global__ void k_rcs(const float* __restrict__ theta, const float* __restrict__ noise,
                      const float* __restrict__ fmask, float* __restrict__ rcs) {
    int idx = blockIdx.x * blockDim.x + threadIdx.x;   // fv*NOUT + n
    if (idx >= F_ * V_ * NOUT_) return;
    int fv = idx / NOUT_, n = idx % NOUT_;
    int f = fv / V_;
    const float* nz = noise + (size_t)fv * M_ * NOUT_ + n;
    const float* fz = fmask + (size_t)f * M_ * NOUT_ + n;
    float s = 0.f;
    for (int m = 0; m < M_; ++m) {
        float tc = clampG(theta[(size_t)m * NOUT_ + n]);
        float tz = (fabsf(tc) < GMIN_) ? 0.f : tc;
        float nf = nz[(size_t)m * NOUT_] * 0.2f + 0.9f;
        s += fabsf(tz * nf * fz[(size_t)m * NOUT_]);
    }
    rcs[idx] = 1.f / (s + 1e-10f);       // IEEE divide, once per (f,v,n)
}

// ---------- K5: AAv[f,m]=sum_{v,e} ax^2 , BBv[f,m]=sum_{v,e} an^2 ----------
__global__ void k_aabb(const float* __restrict__ a, const float* __restrict__ betasel,
                       float* __restrict__ AAv, float* __restrict__ BBv) {
    int idx = blockIdx.x * blockDim.x + threadIdx.x;
    if (idx >= F_ * M_) return;
    int f = idx / M_, m = idx % M_;
    float bs = (m < M_ - 1) ? betasel[m] : 0.f;
    float aa = 0.f, bb = 0.f;
    for (int v = 0; v < V_; ++v) {
        const float* ap = a + ((size_t)(f * V_ + v) * E_) * NIN_;
        for (int e = 0; e < E_; ++e) {
            float x = (m < NIN_) ? ap[(size_t)e * NIN_ + m] : ((m == NIN_) ? 1.f : 0.f);
            float an = (m >= NIN_ + 1) ? 0.f : inv_fn(x, bs);
            aa += x * x;
            bb += an * an;
        }
    }
    AAv[idx] = aa;
    BBv[idx] = bb;
}

// ---------- K4: Pm[f,m]=sum_n gt*p , Qm[f,m]=sum_n gt*(1-p); p=(tz*fm>=0) ----------
__global__ void k_pq(const float* __restrict__ theta, const float* __restrict__ fmask,
                     const float* __restrict__ pg, float* __restrict__ Pm,
                     float* __restrict__ Qm) {
    int b = blockIdx.x;                 // b = f*M_ + m
    int f = b / M_, m = b % M_;
    int tid = threadIdx.x;
    const float* th = theta + (size_t)m * NOUT_;
    const float* fm = fmask + ((size_t)f * M_ + m) * NOUT_;
    float sp = 0.f, sq = 0.f;
    for (int n = tid; n < NOUT_; n += blockDim.x) {
        float tc = clampG(th[n]);
        float gi = fabsf(tc);
        float tz = (gi < GMIN_) ? 0.f : tc;
        float gt = gi * pg[n];
        if (tz * fm[n] >= 0.f) sp += gt; else sq += gt;
    }
    __shared__ float rp[256], rq[256];
    rp[tid] = sp; rq[tid] = sq;
    __syncthreads();
    for (int s = 128; s > 0; s >>= 1) {
        if (tid < s) { rp[tid] += rp[tid + s]; rq[tid] += rq[tid + s]; }
        __syncthreads();
    }
    if (tid == 0) { Pm[b] = rp[0]; Qm[b] = rq[0]; }
}

// ---------- K3: fused dual-GEMM (z and S) with V_WMMA_F32_16X16X4_F32 ----------
// grid: 128 blocks = 16 (f,v) x 8 n-groups; block = 256 thr = 8 waves.
// wave w -> 16-column n-tile; computes full E=64 strip: 4 z-subtiles + 4 S-subtiles.
__global__ __launch_bounds__(256) void k_main(
    const float* __restrict__ a, const float* __restrict__ theta,
    const float* __restrict__ noise, const float* __restrict__ fmask,
    const float* __restrict__ betasel, const float* __restrict__ actsel,
    const float* __restrict__ pg, const float* __restrict__ gtsum,
    const float* __restrict__ rcs, float* __restrict__ out,
    float* __restrict__ pwpart) {
    __shared__ float axl[E_ * 66];   // pad 66 (even -> aligned float2, few bank conflicts)
    __shared__ float anl[E_ * 66];
    __shared__ float wred[8];

    int fv = blockIdx.x >> 3;
    int ngrp = blockIdx.x & 7;
    int f = fv >> 2;
    int tid = threadIdx.x;
    int w = tid >> 5, lane = tid & 31;
    int half = lane >> 4, lr = lane & 15;
    int n = ngrp * 128 + w * 16 + lr;     // this lane's fixed B/C column

    const float* abase = a + (size_t)fv * E_ * NIN_;
    const float* nbase = noise + (size_t)fv * M_ * NOUT_ + n;
    const float* fbase = fmask + (size_t)f * M_ * NOUT_ + n;
    const float* tbase = theta + n;
    float pg_n  = pg[n];
    float rcs_n = rcs[fv * NOUT_ + n];

    const v8f vzero = {0.f, 0.f, 0.f, 0.f, 0.f, 0.f, 0.f, 0.f};
    v8f zacc[4], sacc[4];
#pragma unroll
    for (int t = 0; t < 4; ++t) { zacc[t] = vzero; sacc[t] = vzero; }

    for (int m0 = 0; m0 < M_; m0 += 64) {
        __syncthreads();
        // Stage ax and an (bias-one / zero columns, sigmoid-negation via hw rcp) into LDS
        for (int i = tid; i < E_ * 64; i += 256) {
            int e = i >> 6, mo = i & 63, m = m0 + mo;
            float x = (m < NIN_) ? abase[(size_t)e * NIN_ + m]
                                 : ((m == NIN_) ? 1.f : 0.f);
            float an = (m >= NIN_ + 1) ? 0.f : inv_fn(x, betasel[m]);
            axl[e * 66 + mo] = x;
            anl[e * 66 + mo] = an;
        }
        __syncthreads();

        int rem = M_ - m0;
        int ksteps = (rem < 64) ? ((rem + 3) >> 2) : 16;

        // Software pipeline: prefetch B-inputs (theta, noise, fmask) one K-step ahead
        float thP[2], uP[2], fmP[2];
        {
            int mB0 = m0 + 2 * half;
#pragma unroll
            for (int i = 0; i < 2; ++i) {
                int m = mB0 + i;
                size_t mc = (size_t)((m < M_) ? m : 0) * NOUT_;
                thP[i] = tbase[mc];
                uP[i]  = nbase[mc];
                fmP[i] = fbase[mc];
            }
        }
        for (int ks = 0; ks < ksteps; ++ks) {
            int moff = ks * 4 + 2 * half;
            int mB0 = m0 + moff;
            float thC[2], uC[2], fmC[2];
#pragma unroll
            for (int i = 0; i < 2; ++i) { thC[i] = thP[i]; uC[i] = uP[i]; fmC[i] = fmP[i]; }
            {   // issue next step's loads now (clamped; redundant on last step)
                int ksn = (ks + 1 < ksteps) ? (ks + 1) : ks;
                int mN0 = m0 + ksn * 4 + 2 * half;
#pragma unroll
                for (int i = 0; i < 2; ++i) {
                    int m = mN0 + i;
                    size_t mc = (size_t)((m < M_) ? m : 0) * NOUT_;
                    thP[i] = tbase[mc];
                    uP[i]  = nbase[mc];
                    fmP[i] = fbase[mc];
                }
            }
            // Build the two B-weight pairs for z (Wp/Wn) and S (Gp/Gn)
            v2f Wp, Wn, Gp, Gn;
#pragma unroll
            for (int i = 0; i < 2; ++i) {
                bool valid = (mB0 + i < M_);
                float tc = clampG(thC[i]);
                float gi = fabsf(tc);
                float tz = (gi < GMIN_) ? 0.f : tc;
                float gt = gi * pg_n;
                float nf = uC[i] * 0.2f + 0.9f;
                float tn = tz * nf * fmC[i];
                bool pos = (tn >= 0.f);
                float wv = fabsf(tn) * rcs_n;
                Wp[i] = (valid && pos) ? wv : 0.f;
                Wn[i] = (valid && !pos) ? wv : 0.f;
                Gp[i] = (valid && pos) ? gt : 0.f;
                Gn[i] = (valid && !pos) ? gt : 0.f;
            }
#pragma unroll
            for (int t = 0; t < 4; ++t) {
                int e = t * 16 + lr;
                const float* pax = &axl[e * 66 + moff];
                const float* pan = &anl[e * 66 + moff];
                v2f Af; Af[0] = pax[0]; Af[1] = pax[1];
                v2f An; An[0] = pan[0]; An[1] = pan[1];
                zacc[t] = __builtin_amdgcn_wmma_f32_16x16x4_f32(
                    false, Af, false, Wp, (short)0, zacc[t], false, false);
                zacc[t] = __builtin_amdgcn_wmma_f32_16x16x4_f32(
                    false, An, false, Wn, (short)0, zacc[t], false, false);
                sacc[t] = __builtin_amdgcn_wmma_f32_16x16x4_f32(
                    false, Af, false, Gp, (short)0, sacc[t], false, false);
                sacc[t] = __builtin_amdgcn_wmma_f32_16x16x4_f32(
                    false, An, false, Gn, (short)0, sacc[t], false, false);
            }
        }
    }

    // Epilogue: activation output + power partial  z*(gtsum[n]*z - 2*S)
    float gs_n = gtsum[n];
    float asel = actsel[n];
    float pw = 0.f;
    float* obase = out + (size_t)fv * E_ * NOUT_ + n;
#pragma unroll
    for (int t = 0; t < 4; ++t) {
#pragma unroll
        for (int i = 0; i < 8; ++i) {
            int e = t * 16 + half * 8 + i;
            float z = zacc[t][i], s = sacc[t][i];
            pw += z * (gs_n * z - 2.f * s);
            // branchless: ACT0 = sigmoid(4z-2), ACT1 = tanh(z); both via hw exp+rcp
            float a0 = fsigm(4.f * z - 2.f);
            float a1 = ftanh(z);
            obase[(size_t)e * NOUT_] = (asel > 0.5f) ? a1 : a0;
        }
    }
    for (int off = 16; off; off >>= 1) pw += __shfl_xor(pw, off, 32);
    if (lane == 0) wred[w] = pw;
    __syncthreads();
    if (tid == 0) {
        float s = 0.f;
#pragma unroll
        for (int i = 0; i < 8; ++i) s += wred[i];
        pwpart[blockIdx.x] = s;   // deterministic per-block partial
    }
}

// ---------- K6: final deterministic reduction -> power scalar ----------
__global__ void k_final(const float* __restrict__ Pm, const float* __restrict__ Qm,
                        const float* __restrict__ AAv, const float* __restrict__ BBv,
                        const float* __restrict__ pwpart, float* __restrict__ outp) {
    __shared__ float red[256];
    int tid = threadIdx.x;
    float s = 0.f;
    for (int i = tid; i < F_ * M_; i += 256) s += Pm[i] * AAv[i] + Qm[i] * BBv[i];
    for (int i = tid; i < 128; i += 256) s += pwpart[i];
    red[tid] = s;
    __syncthreads();
    for (int st = 128; st > 0; st >>= 1) {
        if (tid < st) red[tid] += red[tid + st];
        __syncthreads();
    }
    if (tid == 0) outp[0] = red[0] / (float)(E_ * V_ * F_);   // / 1024
}

extern "C" void kernel_launch(void* const* d_in, const int* in_sizes, int n_in,
                              void* d_out, int out_size, void* d_ws, size_t ws_size,
                              hipStream_t stream) {
    const float* a         = (const float*)d_in[0];
    const float* theta     = (const float*)d_in[1];
    const float* coeff_act = (const float*)d_in[2];
    const float* coeff_neg = (const float*)d_in[3];
    const float* noise     = (const float*)d_in[4];
    const float* gum_act   = (const float*)d_in[5];
    const float* gum_neg   = (const float*)d_in[6];
    const float* fmask     = (const float*)d_in[7];
    float* out = (float*)d_out;

    float* ws = (float*)d_ws;
    float* pg      = ws; ws += NOUT_;
    float* gtsum   = ws; ws += NOUT_;
    float* actsel  = ws; ws += NOUT_;
    float* betasel = ws; ws += M_ + 2;          // keep alignment
    float* rcs     = ws; ws += F_ * V_ * NOUT_;
    float* Pm      = ws; ws += F_ * M_;
    float* Qm      = ws; ws += F_ * M_;
    float* AAv     = ws; ws += F_ * M_;
    float* BBv     = ws; ws += F_ * M_;
    float* pwpart  = ws; ws += 128;

    k_col <<<NOUT_ / 256, 256, 0, stream>>>(theta, coeff_act, gum_act, pg, gtsum, actsel);
    k_beta<<<(M_ + 255) / 256, 256, 0, stream>>>(coeff_neg, gum_neg, betasel);
    k_rcs <<<(F_ * V_ * NOUT_) / 256, 256, 0, stream>>>(theta, noise, fmask, rcs);
    k_aabb<<<(F_ * M_ + 255) / 256, 256, 0, stream>>>(a, betasel, AAv, BBv);
    k_pq  <<<F_ * M_, 256, 0, stream>>>(theta, fmask, pg, Pm, Qm);
    k_main<<<128, 256, 0, stream>>>(a, theta, noise, fmask, betasel, actsel,
                                    pg, gtsum, rcs, out, pwpart);
    k_final<<<1, 256, 0, stream>>>(Pm, Qm, AAv, BBv, pwpart, out + (out_size - 1));
}